// GLE_TS_4647154614867
// MI455X (gfx1250) — compile-verified
//
#include <hip/hip_runtime.h>
#include <hip/hip_bf16.h>

#define GLE_BOLTZMAN 0.001987191f
#define GLE_L 64

typedef __attribute__((ext_vector_type(2))) float v2f;
typedef __attribute__((ext_vector_type(4))) float v4f;
typedef __attribute__((ext_vector_type(8))) float v8f;

// ---------------------------------------------------------------------------
// Prep kernel (1 block, 64 threads): build shifted tap kernels in d_ws.
//   ws[0..63]    kv[s] = (s==0) ? 0 : -mk[64-s]      (mk = scaled autocorr of h)
//   ws[64..127]  kw[s] = (s==0) ? 0 :  h[64-s]
//   ws[128]      cv    = -mk[0]   (coefficient for newest sample v)
//   ws[129]      cw    =  h[0]    (coefficient for newest sample w_new)
// ---------------------------------------------------------------------------
__global__ void gle_prep_kernel(const float* __restrict__ h,
                                const float* __restrict__ mass,
                                const int* __restrict__ T,
                                float* __restrict__ ws) {
    __shared__ float sh[GLE_L];
    __shared__ float smk[GLE_L];
    const int k = threadIdx.x;
    sh[k] = h[k];
    __syncthreads();
    float acc = 0.f;
    for (int t = 0; t + k < GLE_L; ++t) acc += sh[t] * sh[t + k];
    const float scale = mass[0] * GLE_BOLTZMAN * (float)T[0];
    smk[k] = acc * scale;
    __syncthreads();
    ws[k]         = (k == 0) ? 0.f : -smk[GLE_L - k];
    ws[GLE_L + k] = (k == 0) ? 0.f :  sh[GLE_L - k];
    if (k == 0) {
        ws[2 * GLE_L]     = -smk[0];
        ws[2 * GLE_L + 1] =  sh[0];
    }
}

// ---------------------------------------------------------------------------
// Main kernel: one wave32 per 16-channel tile.
// A = 16x4 f32 slice of the data rows, B = tap weights replicated across N,
// C accumulates D[m][n] = dot(row m, taps) (identical across n).
// 16 K-steps x 2 WMMAs (v-path + w-path) per tile.
// ---------------------------------------------------------------------------
__global__ void gle_wmma_kernel(const float* __restrict__ vlist,
                                const float* __restrict__ wlist,
                                const float* __restrict__ v,
                                const float* __restrict__ wnew,
                                const float* __restrict__ ws,
                                float* __restrict__ out,
                                int ntiles) {
    const int lane = threadIdx.x & 31;
    const int tile = blockIdx.x * (blockDim.x >> 5) + (threadIdx.x >> 5);
    if (tile >= ntiles) return;           // wave-uniform: EXEC stays all-ones

    const int rowBase = tile * 16;
    const int m       = lane & 15;        // matrix row owned by this lane
    const int khalf   = (lane >> 4) << 1; // 0 for lanes 0-15, 2 for lanes 16-31

    const float* __restrict__ vrow = vlist + (size_t)(rowBase + m) * GLE_L;
    const float* __restrict__ wrow = wlist + (size_t)(rowBase + m) * GLE_L;

    v8f acc = {};
#pragma unroll
    for (int kb = 0; kb < GLE_L; kb += 4) {
        const int kk = kb + khalf;
        v2f av = *(const v2f*)(vrow + kk);
        v2f aw = *(const v2f*)(wrow + kk);
        v2f bv = { ws[kk],         ws[kk + 1] };
        v2f bw = { ws[GLE_L + kk], ws[GLE_L + kk + 1] };
        acc = __builtin_amdgcn_wmma_f32_16x16x4_f32(false, av, false, bv,
                                                    (short)0, acc, false, false);
        acc = __builtin_amdgcn_wmma_f32_16x16x4_f32(false, aw, false, bw,
                                                    (short)0, acc, false, false);
    }

    // D layout: lanes 0-15 hold channels rowBase+0..7 in acc[0..7],
    //           lanes 16-31 hold channels rowBase+8..15.
    if (m == 0) {
        const float cv = ws[2 * GLE_L];
        const float cw = ws[2 * GLE_L + 1];
        const int base = rowBase + ((lane >> 4) << 3);
        float r[8];
#pragma unroll
        for (int j = 0; j < 8; ++j)
            r[j] = acc[j] + cv * v[base + j] + cw * wnew[base + j];
        *(v4f*)(out + base)     = (v4f){r[0], r[1], r[2], r[3]};
        *(v4f*)(out + base + 4) = (v4f){r[4], r[5], r[6], r[7]};
    }
}

// ---------------------------------------------------------------------------
// Scalar tail kernel (only launched if channel count % 16 != 0; with
// NATOM=200000 it never runs, but keeps the code shape-robust).
// ---------------------------------------------------------------------------
__global__ void gle_tail_kernel(const float* __restrict__ vlist,
                                const float* __restrict__ wlist,
                                const float* __restrict__ v,
                                const float* __restrict__ wnew,
                                const float* __restrict__ ws,
                                float* __restrict__ out,
                                int start, int n) {
    int i = start + blockIdx.x * blockDim.x + threadIdx.x;
    if (i >= n) return;
    const float* vr = vlist + (size_t)i * GLE_L;
    const float* wr = wlist + (size_t)i * GLE_L;
    float acc = 0.f;
    for (int s = 0; s < GLE_L; ++s)
        acc += vr[s] * ws[s] + wr[s] * ws[GLE_L + s];
    out[i] = acc + ws[2 * GLE_L] * v[i] + ws[2 * GLE_L + 1] * wnew[i];
}

extern "C" void kernel_launch(void* const* d_in, const int* in_sizes, int n_in,
                              void* d_out, int out_size, void* d_ws, size_t ws_size,
                              hipStream_t stream) {
    // setup_inputs order: v, T, dt, mass, h, v_list, w_list, w_new
    const float* v     = (const float*)d_in[0];
    const int*   T     = (const int*)  d_in[1];
    // d_in[2] = dt (unused)
    const float* mass  = (const float*)d_in[3];
    const float* h     = (const float*)d_in[4];
    const float* vlist = (const float*)d_in[5];
    const float* wlist = (const float*)d_in[6];
    const float* wnew  = (const float*)d_in[7];
    float*       out   = (float*)d_out;
    float*       ws    = (float*)d_ws;

    const int n = in_sizes[0];            // 3N flat channels (600000)
    const int ntiles = n / 16;
    const int tail   = n - ntiles * 16;

    gle_prep_kernel<<<1, GLE_L, 0, stream>>>(h, mass, T, ws);

    if (ntiles > 0) {
        const int wavesPerBlock = 8;      // 256 threads = 8 wave32
        const int blocks = (ntiles + wavesPerBlock - 1) / wavesPerBlock;
        gle_wmma_kernel<<<blocks, wavesPerBlock * 32, 0, stream>>>(
            vlist, wlist, v, wnew, ws, out, ntiles);
    }
    if (tail > 0) {
        gle_tail_kernel<<<(tail + 63) / 64, 64, 0, stream>>>(
            vlist, wlist, v, wnew, ws, out, ntiles * 16, n);
    }
}